// Yolo_loss_44126493999482
// MI455X (gfx1250) — compile-verified
//
#include <hip/hip_runtime.h>
#include <hip/hip_bf16.h>
#include <stdint.h>

// ---------------- constants ----------------
#define NCLS 80
#define NCH 85
#define TMAX 60
#define BATCH 16
#define LOG_EPS 3.72e-44f          // matches torch BCE log clamp exp(-100)
#define FOUR_OVER_PI2 0.40528473456935109f

__device__ __constant__ float c_anchors[9][2] = {
    {12,16},{19,36},{40,28},{36,75},{76,55},{72,146},{142,110},{192,243},{459,401}};
__device__ __constant__ float c_stride[3] = {8.f, 16.f, 32.f};
__device__ __constant__ int   c_fs[3]     = {76, 38, 19};

// meta layout per (oid,b,t): 16 floats (4x float4)
//  f4[0] = {txl, tyl, txr, tyr}   half-extent truth box, grid units
//  f4[1] = {areaT, mkey, 0, 0}    mkey = (best_n*Fs+tj)*Fs+ti if matched else -1
//  f4[2] = {fx, fy, lw, lh}       regression targets (matched cells only)
//  f4[3] = {scale, cls, 0, 0}

__device__ inline float logc(float p) {
    return __logf(fminf(fmaxf(p, LOG_EPS), 1.0f));
}
__device__ inline float bce(float p, float t) {
    return -(t * logc(p) + (1.f - t) * logc(1.f - p));
}
__device__ inline float sigm(float x) { return 1.f / (1.f + __expf(-x)); }

// ---------------- Tensor Data Mover staging (gfx1250 TDM) ----------------
// HIP compiles this TU twice. The host pass has no amdgcn builtins, so the
// TDM path must be preprocessor-guarded; the fallback below exists only so
// the host pass can parse the kernel. The #error is device-pass-only: if
// this file compiles, the gfx1250 bundle definitely contains
// tensor_load_to_lds + s_wait_tensorcnt.
#if __has_builtin(__builtin_amdgcn_tensor_load_to_lds) && \
    __has_builtin(__builtin_amdgcn_s_wait_tensorcnt)
#define HAVE_TDM 1
#endif

#if defined(__HIP_DEVICE_COMPILE__) && !defined(HAVE_TDM)
#error "gfx1250 device pass lacks TDM builtins (tensor_load_to_lds / s_wait_tensorcnt)"
#endif

typedef unsigned int u32x4 __attribute__((ext_vector_type(4)));
typedef int          i32x4 __attribute__((ext_vector_type(4)));
typedef int          i32x8 __attribute__((ext_vector_type(8)));

#ifdef HAVE_TDM
// 2D tile load: tile_d0 (X elems, 4B each) x tile_d1 rows, row stride = stride0 elems.
// OOB in X (x >= tensor_d0) zero-fills the LDS destination (handles ragged tail).
__device__ inline void tdm_load_2d(unsigned lds_addr, const void* gptr,
                                   unsigned tensor_d0, unsigned tensor_d1,
                                   unsigned tile_d0, unsigned tile_d1,
                                   unsigned long long stride0_elems)
{
    unsigned long long ga = (unsigned long long)gptr;
    u32x4 g0 = (u32x4)0;
    g0.x = 1u;                                                 // count=1 valid descriptor
    g0.y = lds_addr;                                           // lds_addr [63:32]
    g0.z = (unsigned)(ga & 0xffffffffull);                     // global_addr[31:0]
    g0.w = (unsigned)((ga >> 32) & 0x1ffffffull) | (2u << 30); // addr[56:32] | type=2

    i32x8 g1 = (i32x8)0;
    g1[0] = (int)(2u << 16);                                   // wg_mask=0, data_size=2 (4B)
    g1[1] = (int)((tensor_d0 & 0xffffu) << 16);                // tensor_dim0[15:0] @ [63:48]
    g1[2] = (int)((tensor_d0 >> 16) | ((tensor_d1 & 0xffffu) << 16));
    g1[3] = (int)((tensor_d1 >> 16) | ((tile_d0 & 0xffffu) << 16)); // tile_dim0
    g1[4] = (int)(tile_d1 & 0xffffu);                          // tile_dim1 (tile_dim2=0)
    g1[5] = (int)(stride0_elems & 0xffffffffull);              // tensor_dim0_stride[31:0]
    g1[6] = (int)((stride0_elems >> 32) & 0xffffull);          // stride[47:32]
    g1[7] = 0;                                                 // dim1 stride unused (2D)

    i32x4 z4 = (i32x4)0;
#if defined(__clang_major__) && (__clang_major__ >= 23)
    __builtin_amdgcn_tensor_load_to_lds(g0, g1, z4, z4, (i32x8)0, 0);
#else
    __builtin_amdgcn_tensor_load_to_lds(g0, g1, z4, z4, 0);
#endif
}
#endif // HAVE_TDM

// ---------------- kernel 0: zero accumulators + has_match ----------------
__global__ void yolo_init(float* out, int* hm) {
    int t = threadIdx.x;
    if (t < 6)  out[t] = 0.f;
    if (t < 48) hm[t] = 0;
}

// ---------------- kernel 1: per-(scale,b,t) anchor matching ----------------
__global__ void yolo_prep(const float* __restrict__ labels,
                          float* __restrict__ meta, int* __restrict__ hm)
{
    int tid = blockIdx.x * blockDim.x + threadIdx.x;
    if (tid >= 3 * BATCH * TMAX) return;
    int oid = tid / (BATCH * TMAX);
    int rem = tid % (BATCH * TMAX);
    int b = rem / TMAX, t = rem % TMAX;

    const float* l = labels + (size_t)(b * TMAX + t) * 5;
    float l0 = l[0], l1 = l[1], l2 = l[2], l3 = l[3], l4 = l[4];
    float strd = c_stride[oid];
    int Fs = c_fs[oid];

    bool valid = (l0 + l1 + l2 + l3 + l4) > 0.f;
    float tx = (l2 + l0) / (2.f * strd);
    float ty = (l3 + l1) / (2.f * strd);
    float tw = (l2 - l0) / strd;
    float th = (l3 - l1) / strd;
    int ti = min(max((int)tx, 0), Fs - 1);
    int tj = min(max((int)ty, 0), Fs - 1);
    int cls = min(max((int)l4, 0), NCLS - 1);

    // CIoU of (0,0,tw,th) vs 9 ref anchors (grid units); first-max argmax
    float best = -1e30f; int bestn = 0;
    float areaA = tw * th;
    for (int n = 0; n < 9; ++n) {
        float aw = c_anchors[n][0] / strd, ah = c_anchors[n][1] / strd;
        float brx = fminf(tw, aw), bry = fminf(th, ah);
        float en = (brx > 0.f && bry > 0.f) ? 1.f : 0.f;
        float ai = brx * bry * en;
        float iou = ai / (areaA + aw * ah - ai + 1e-16f);
        float cbx = fmaxf(tw, aw), cby = fmaxf(th, ah);
        float c2 = cbx * cbx + cby * cby + 1e-16f;
        float rho2 = 0.25f * ((tw - aw) * (tw - aw) + (th - ah) * (th - ah));
        float dv = atanf(tw / (th + 1e-16f)) - atanf(aw / (ah + 1e-16f));
        float v = FOUR_OVER_PI2 * dv * dv;
        float alpha = v / (1.f - iou + v + 1e-16f);
        float ciou = iou - (rho2 / c2 + v * alpha);
        if (ciou > best) { best = ciou; bestn = n; }
    }
    int best_n = bestn % 3;
    bool match = valid && (bestn / 3 == oid);

    float aw = c_anchors[oid * 3 + best_n][0] / strd;
    float ah = c_anchors[oid * 3 + best_n][1] / strd;
    float fx = tx - floorf(tx), fy = ty - floorf(ty);
    float lw = logf(tw / aw + 1e-16f);
    float lh = logf(th / ah + 1e-16f);
    float sc = sqrtf(fmaxf(2.f - tw * th / ((float)Fs * (float)Fs), 0.f));
    float mkey = match ? (float)((best_n * Fs + tj) * Fs + ti) : -1.f;

    float4* m = (float4*)(meta + (size_t)((oid * BATCH + b) * TMAX + t) * 16);
    m[0] = make_float4(tx - 0.5f * tw, ty - 0.5f * th,
                       tx + 0.5f * tw, ty + 0.5f * th);
    m[1] = make_float4(tw * th, mkey, 0.f, 0.f);
    m[2] = make_float4(fx, fy, lw, lh);
    m[3] = make_float4(sc, (float)cls, 0.f, 0.f);
    if (match) atomicOr(&hm[oid * BATCH + b], 1);
}

// ---------------- kernel 2: per-cell losses (one scale) ----------------
__global__ void __launch_bounds__(256)
yolo_main(const float* __restrict__ x, const float* __restrict__ meta,
          const int* __restrict__ hm, float* __restrict__ out, int oid)
{
    __shared__ float  ldsX[5 * 256];         // 5 needed channel planes, 256-cell chunk
    __shared__ float4 ldsM4[TMAX * 4];       // packed truth metadata for this (oid,b)

    const int Fs = c_fs[oid];
    const int FsFs = Fs * Fs;
    const int a = blockIdx.y;
    const int b = blockIdx.z;
    const int chunk = blockIdx.x * 256;
    const float strd = c_stride[oid];
    const float* src  = x + (size_t)(b * 3 + a) * NCH * FsFs + chunk;   // ch0 plane @ chunk
    const float* msrc = meta + (size_t)(oid * BATCH + b) * TMAX * 16;

#ifdef HAVE_TDM
    if (threadIdx.x < 32) {   // wave 0 issues both TDM descriptors
        unsigned ldsx_a = (unsigned)(uintptr_t)(void*)ldsX;
        tdm_load_2d(ldsx_a, src, (unsigned)(FsFs - chunk), 5u, 256u, 5u,
                    (unsigned long long)FsFs);
        unsigned ldsm_a = (unsigned)(uintptr_t)(void*)ldsM4;
        tdm_load_2d(ldsm_a, msrc, (unsigned)(TMAX * 16), 1u, (unsigned)(TMAX * 16), 0u,
                    (unsigned long long)(TMAX * 16));
        __builtin_amdgcn_s_wait_tensorcnt(0);
    }
#else
    // host-pass parse stub / non-TDM fallback (device pass never uses this:
    // the device-pass #error above guarantees TDM is taken there)
    for (int c = 0; c < 5; ++c)
        ldsX[c * 256 + threadIdx.x] =
            (chunk + (int)threadIdx.x < FsFs) ? src[(size_t)c * FsFs + threadIdx.x] : 0.f;
    for (int idx = threadIdx.x; idx < TMAX * 16; idx += 256)
        ((float*)ldsM4)[idx] = msrc[idx];
#endif
    __syncthreads();

    float lxy = 0.f, lwh = 0.f, lobj = 0.f, lcls = 0.f, ll2 = 0.f;
    const int cell = chunk + threadIdx.x;
    if (cell < FsFs) {
        int j = cell / Fs, i = cell - j * Fs;
        float o0 = ldsX[0 * 256 + threadIdx.x];
        float o1 = ldsX[1 * 256 + threadIdx.x];
        float o2 = ldsX[2 * 256 + threadIdx.x];
        float o3 = ldsX[3 * 256 + threadIdx.x];
        float o4 = ldsX[4 * 256 + threadIdx.x];
        float s0 = sigm(o0), s1 = sigm(o1), s4 = sigm(o4);
        float aw = c_anchors[oid * 3 + a][0] / strd;
        float ah = c_anchors[oid * 3 + a][1] / strd;
        float px = s0 + (float)i, py = s1 + (float)j;
        float pw = __expf(o2) * aw, ph = __expf(o3) * ah;
        float hx0 = px - 0.5f * pw, hx1 = px + 0.5f * pw;
        float hy0 = py - 0.5f * ph, hy1 = py + 0.5f * ph;
        float pArea = pw * ph;
        const float ck = (float)((a * Fs + j) * Fs + i);   // exact: < 2^24

        // branchless ignore-mask + scatter-match scan over 60 truths.
        // iou > 0.5  <=>  3*ai > pArea + areaT + eps   (denominator positive)
        const float2* m2 = (const float2*)ldsM4;
        bool ign = false;
        int mt = -1;
        for (int t = 0; t < TMAX; ++t) {
            float4 v0 = ldsM4[t * 4 + 0];          // txl,tyl,txr,tyr
            float2 v1 = m2[t * 8 + 2];             // areaT, mkey
            float tlx = fmaxf(hx0, v0.x), tly = fmaxf(hy0, v0.y);
            float brx = fminf(hx1, v0.z), bry = fminf(hy1, v0.w);
            float ai = ((tlx < brx) && (tly < bry)) ? (brx - tlx) * (bry - tly) : 0.f;
            ign = ign || (3.f * ai > pArea + v1.x + 1e-16f);
            if (v1.y == ck) mt = t;                // scatter semantics: last writer wins
        }
        float obj = (hm[oid * BATCH + b] != 0) ? (ign ? 0.f : 1.f) : 1.f;

        if (mt >= 0) {                  // matched cell (scatter forces obj_mask=1)
            float4 r   = ldsM4[mt * 4 + 2];        // fx,fy,lw,lh
            float2 scv = m2[mt * 8 + 6];           // scale, cls
            float w2 = scv.x * scv.x;
            lxy = (bce(s0, r.x) + bce(s1, r.y)) * w2;
            float dw = o2 - r.z, dh = o3 - r.w;
            float whq = dw * dw + dh * dh;
            lwh = 0.5f * w2 * whq;
            lobj = bce(s4, 1.f);
            float d0 = s0 - r.x, d1 = s1 - r.y, d4 = s4 - 1.f;
            ll2 = d0 * d0 + d1 * d1 + w2 * whq + d4 * d4;
            int cls = (int)scv.y;
            const float* xc = x + (size_t)(b * 3 + a) * NCH * FsFs
                                + (size_t)5 * FsFs + cell;
            for (int k = 0; k < NCLS; ++k) {   // sparse: only matched cells read classes
                float sk = sigm(xc[(size_t)k * FsFs]);
                float tk = (k == cls) ? 1.f : 0.f;
                lcls += bce(sk, tk);
                float dk = sk - tk;
                ll2 += dk * dk;
            }
        } else {                        // unmatched: only obj channel contributes
            float p = s4 * obj;
            lobj = bce(p, 0.f);
            ll2 = p * p;
        }
    }

    // wave32 tree reduce, then fp32 global atomics (one lane per wave)
    for (int off = 16; off > 0; off >>= 1) {
        lxy  += __shfl_down(lxy,  off, 32);
        lwh  += __shfl_down(lwh,  off, 32);
        lobj += __shfl_down(lobj, off, 32);
        lcls += __shfl_down(lcls, off, 32);
        ll2  += __shfl_down(ll2,  off, 32);
    }
    if ((threadIdx.x & 31) == 0) {
        atomicAdd(&out[1], lxy);
        atomicAdd(&out[2], lwh);
        atomicAdd(&out[3], lobj);
        atomicAdd(&out[4], lcls);
        atomicAdd(&out[5], ll2);
    }
}

// ---------------- kernel 3: finalize ----------------
__global__ void yolo_fin(float* out) {
    out[0] = out[1] + out[2] + out[3] + out[4];
}

// ---------------- launch ----------------
extern "C" void kernel_launch(void* const* d_in, const int* in_sizes, int n_in,
                              void* d_out, int out_size, void* d_ws, size_t ws_size,
                              hipStream_t stream)
{
    const float* x[3]   = {(const float*)d_in[0], (const float*)d_in[1], (const float*)d_in[2]};
    const float* labels = (const float*)d_in[3];
    float* out  = (float*)d_out;
    float* meta = (float*)d_ws;                                   // 3*16*60*16 floats
    int*   hm   = (int*)((char*)d_ws + (size_t)3 * BATCH * TMAX * 16 * sizeof(float));

    yolo_init<<<1, 64, 0, stream>>>(out, hm);

    int np = 3 * BATCH * TMAX;
    yolo_prep<<<(np + 255) / 256, 256, 0, stream>>>(labels, meta, hm);

    const int fs[3] = {76, 38, 19};
    for (int oid = 0; oid < 3; ++oid) {
        int FsFs = fs[oid] * fs[oid];
        dim3 grid((FsFs + 255) / 256, 3, BATCH);
        yolo_main<<<grid, 256, 0, stream>>>(x[oid], meta, hm, out, oid);
    }
    yolo_fin<<<1, 1, 0, stream>>>(out);
}